// SplitGrid_54254026883541
// MI455X (gfx1250) — compile-verified
//
#include <hip/hip_runtime.h>

// ---------------------------------------------------------------------------
// SplitGrid MoE: 1M points -> 64 experts (8x8 grid), per-expert 2->32->16 MLP.
// Bin points by expert, then layer-2 as dense 16x16 f32 WMMA tiles
// (8 x V_WMMA_F32_16X16X4_F32 per tile, K=32).
// ---------------------------------------------------------------------------

typedef float v2f __attribute__((ext_vector_type(2)));
typedef float v8f __attribute__((ext_vector_type(8)));

__device__ __forceinline__ int expert_of(float x, float y) {
    int i = (int)floorf(x * 8.0f);
    int j = (int)floorf(y * 8.0f);
    i = i < 0 ? 0 : (i > 7 ? 7 : i);
    j = j < 0 ? 0 : (j > 7 ? 7 : j);
    return j * 8 + i;
}

__global__ void k_init(int* __restrict__ counts) {
    counts[threadIdx.x] = 0;
}

__global__ void k_count(const float2* __restrict__ s, int* __restrict__ counts, int n) {
    __shared__ int hist[64];
    if (threadIdx.x < 64) hist[threadIdx.x] = 0;
    __syncthreads();
    for (int p = blockIdx.x * blockDim.x + threadIdx.x; p < n;
         p += gridDim.x * blockDim.x) {
        float2 xy = s[p];
        atomicAdd(&hist[expert_of(xy.x, xy.y)], 1);
    }
    __syncthreads();
    if (threadIdx.x < 64) atomicAdd(&counts[threadIdx.x], hist[threadIdx.x]);
}

__global__ void k_scan(const int* __restrict__ counts, int* __restrict__ offsets,
                       int* __restrict__ cursors) {
    int acc = 0;
    for (int e = 0; e < 64; ++e) {
        offsets[e] = acc;
        cursors[e] = acc;
        acc += counts[e];
    }
}

__global__ __launch_bounds__(1024) void k_scatter(const float2* __restrict__ s,
                                                  int* __restrict__ cursors,
                                                  int* __restrict__ idx, int n) {
    __shared__ int lcnt[64];
    __shared__ int lbase[64];
    int t = threadIdx.x;
    if (t < 64) lcnt[t] = 0;
    __syncthreads();
    int p = blockIdx.x * blockDim.x + t;
    int e = 0, lrank = 0;
    bool ok = (p < n);
    if (ok) {
        float2 xy = s[p];
        e = expert_of(xy.x, xy.y);
        lrank = atomicAdd(&lcnt[e], 1);
    }
    __syncthreads();
    if (t < 64 && lcnt[t] > 0) lbase[t] = atomicAdd(&cursors[t], lcnt[t]);
    __syncthreads();
    if (ok) idx[lbase[e] + lrank] = p;
}

// One expert per blockIdx.x; blockIdx.y slices the expert's tile stream.
// 8 waves/block, each wave owns a 16-point tile per iteration.
#define SSTRIDE 20   // LDS row stride (floats): 16B-aligned, conflict-free halves

__global__ __launch_bounds__(256) void k_mlp(
    const float2* __restrict__ samples,
    const float*  __restrict__ W1,   // [E][2][32]
    const float*  __restrict__ B1,   // [E][32]
    const float*  __restrict__ W2,   // [E][32][16]
    const float*  __restrict__ B2,   // [E][16]
    const int*    __restrict__ counts,
    const int*    __restrict__ offsets,
    const int*    __restrict__ idx,
    float*        __restrict__ out)  // [N][16]
{
    __shared__ float sW1[64];
    __shared__ float sB1[32];
    __shared__ float sW2[512];
    __shared__ float sB2[16];
    __shared__ float stage[8][16 * SSTRIDE];

    const int e   = blockIdx.x;
    const int tid = threadIdx.x;

    if (tid < 64)               sW1[tid]      = W1[e * 64 + tid];
    if (tid >= 64 && tid < 96)  sB1[tid - 64] = B1[e * 32 + (tid - 64)];
    if (tid >= 96 && tid < 112) sB2[tid - 96] = B2[e * 16 + (tid - 96)];
    for (int i = tid; i < 512; i += 256) sW2[i] = W2[e * 512 + i];
    __syncthreads();

    const int lane = tid & 31;
    const int wave = tid >> 5;
    const int half = lane >> 4;   // 0: K%4 in {0,1}; 1: K%4 in {2,3}
    const int L    = lane & 15;   // row (A) / column (B,C,D) within tile

    // W2 in WMMA B-layout registers, reused for every tile:
    // B chunk k = rows 4k..4k+3; lane half h holds rows 4k+2h, 4k+2h+1, col L.
    v2f Breg[8];
#pragma unroll
    for (int k = 0; k < 8; ++k) {
        int row = 4 * k + 2 * half;
        Breg[k].x = sW2[row * 16 + L];
        Breg[k].y = sW2[(row + 1) * 16 + L];
    }
    const float bias2 = sB2[L];
    const int   cnt   = counts[e];
    const int   base  = offsets[e];
    const int   nfull = cnt >> 4;       // full 16-point tiles
    const int   rem   = cnt & 15;

    float* st = &stage[wave][0];
    int t = blockIdx.y * 8 + wave;

    // ---------- fast path: full tiles, no predication ----------
    for (; t < nfull; t += 64) {
        int p     = idx[base + t * 16 + L];
        float2 xy = samples[p];

        v2f a[8];
#pragma unroll
        for (int k = 0; k < 8; ++k) {
            int h0 = 4 * k + 2 * half;
            float v0 = fmaf(xy.x, sW1[h0],     fmaf(xy.y, sW1[32 + h0],     sB1[h0]));
            float v1 = fmaf(xy.x, sW1[h0 + 1], fmaf(xy.y, sW1[32 + h0 + 1], sB1[h0 + 1]));
            a[k].x = fmaxf(v0, 0.0f);
            a[k].y = fmaxf(v1, 0.0f);
        }

        v8f c;
#pragma unroll
        for (int r = 0; r < 8; ++r) c[r] = bias2;
#pragma unroll
        for (int k = 0; k < 8; ++k)
            c = __builtin_amdgcn_wmma_f32_16x16x4_f32(
                    false, a[k], false, Breg[k], (short)0, c, false, false);

        // Transpose D through per-wave LDS: element [M=8*half+r][N=L].
#pragma unroll
        for (int r = 0; r < 8; ++r) st[(8 * half + r) * SSTRIDE + L] = c[r];

        // Each lane flushes half of one output row as two b128 stores.
        int mr = lane >> 1;             // tile row 0..15
        int q  = lane & 1;              // which half of the row
        int pm = __shfl(p, mr, 32);     // point index of tile row mr
        float4 v0 = *(const float4*)&st[mr * SSTRIDE + q * 8];
        float4 v1 = *(const float4*)&st[mr * SSTRIDE + q * 8 + 4];
        float4* orow = (float4*)(out + (size_t)pm * 16 + q * 8);
        orow[0] = v0;
        orow[1] = v1;
    }

    // ---------- remainder tile: exactly one wave (t == nfull) ----------
    if (rem && t == nfull) {
        int slot    = nfull * 16 + L;
        int clamped = slot < cnt ? slot : cnt - 1;   // pad with last point
        int p       = idx[base + clamped];
        float2 xy   = samples[p];

        v2f a[8];
#pragma unroll
        for (int k = 0; k < 8; ++k) {
            int h0 = 4 * k + 2 * half;
            float v0 = fmaf(xy.x, sW1[h0],     fmaf(xy.y, sW1[32 + h0],     sB1[h0]));
            float v1 = fmaf(xy.x, sW1[h0 + 1], fmaf(xy.y, sW1[32 + h0 + 1], sB1[h0 + 1]));
            a[k].x = fmaxf(v0, 0.0f);
            a[k].y = fmaxf(v1, 0.0f);
        }

        v8f c;
#pragma unroll
        for (int r = 0; r < 8; ++r) c[r] = bias2;
#pragma unroll
        for (int k = 0; k < 8; ++k)
            c = __builtin_amdgcn_wmma_f32_16x16x4_f32(
                    false, a[k], false, Breg[k], (short)0, c, false, false);

#pragma unroll
        for (int r = 0; r < 8; ++r) {
            int m  = half * 8 + r;
            int pm = __shfl(p, m, 32);
            if (nfull * 16 + m < cnt) out[(size_t)pm * 16 + L] = c[r];
        }
    }
}

extern "C" void kernel_launch(void* const* d_in, const int* in_sizes, int n_in,
                              void* d_out, int out_size, void* d_ws, size_t ws_size,
                              hipStream_t stream) {
    const float2* samples = (const float2*)d_in[0];
    const float*  W1      = (const float*)d_in[1];
    const float*  B1      = (const float*)d_in[2];
    const float*  W2      = (const float*)d_in[3];
    const float*  B2      = (const float*)d_in[4];
    float*        out     = (float*)d_out;

    const int n = in_sizes[0] / 2;

    int* counts  = (int*)d_ws;       // [64]
    int* offsets = counts + 64;      // [64]
    int* cursors = counts + 128;     // [64]
    int* idx     = counts + 256;     // [n]

    k_init<<<1, 64, 0, stream>>>(counts);
    k_count<<<512, 256, 0, stream>>>(samples, counts, n);
    k_scan<<<1, 1, 0, stream>>>(counts, offsets, cursors);
    k_scatter<<<(n + 1023) / 1024, 1024, 0, stream>>>(samples, cursors, idx, n);
    k_mlp<<<dim3(64, 8), 256, 0, stream>>>(samples, W1, B1, W2, B2,
                                           counts, offsets, idx, out);
}